// topkCE_22479858827720
// MI455X (gfx1250) — compile-verified
//
#include <hip/hip_runtime.h>
#include <hip/hip_bf16.h>

#define C_DIM   32000
#define N_ROWS  8192
#define KCOUNT  5734            // int(0.7 * 8192)
#define BLK1    256
#define BLK2    1024

typedef float f32x4 __attribute__((ext_vector_type(4)));
typedef int   i32x4 __attribute__((ext_vector_type(4)));

__device__ __forceinline__ float wave_max32(float v) {
#pragma unroll
  for (int off = 16; off > 0; off >>= 1)
    v = fmaxf(v, __shfl_xor(v, off, 32));
  return v;
}

__device__ __forceinline__ float wave_sum32(float v) {
#pragma unroll
  for (int off = 16; off > 0; off >>= 1)
    v += __shfl_xor(v, off, 32);
  return v;
}

// ---------------------------------------------------------------------------
// Kernel 1: one workgroup per row. Stage the 128 KB row into LDS with the
// CDNA5 async data-mover path (GLOBAL_LOAD_ASYNC_TO_LDS_B128, ASYNCcnt),
// then compute loss = rowmax + log(sum exp(x - rowmax)) - x[target].
// ---------------------------------------------------------------------------
__global__ void __launch_bounds__(BLK1)
row_nll_kernel(const float* __restrict__ inp, const int* __restrict__ tgt,
               float* __restrict__ losses) {
  extern __shared__ __align__(16) float row[];   // C_DIM floats = 128000 B
  __shared__ float red[16];

  const int rowid = blockIdx.x;
  const int tid   = threadIdx.x;
  const int lane  = tid & 31;
  const int wave  = tid >> 5;
  const float* g  = inp + (size_t)rowid * C_DIM;

  // ---- stage global row -> LDS ----
#if defined(__gfx1250__) && __has_builtin(__builtin_amdgcn_global_load_async_to_lds_b128)
  {
    typedef __attribute__((address_space(1))) i32x4* gvec_t;
    typedef __attribute__((address_space(3))) i32x4* lvec_t;
    gvec_t gp = (gvec_t)g;          // global int4 pointer (const dropped)
    lvec_t lp = (lvec_t)row;        // LDS int4 pointer
#pragma unroll 4
    for (int i = tid; i < C_DIM / 4; i += BLK1) {
      __builtin_amdgcn_global_load_async_to_lds_b128(gp + i, lp + i,
                                                     /*offset=*/0, /*cpol=*/0);
    }
#if __has_builtin(__builtin_amdgcn_s_wait_asynccnt)
    __builtin_amdgcn_s_wait_asynccnt(0);
#else
    asm volatile("s_wait_asynccnt 0" ::: "memory");
#endif
  }
#else
#pragma message("CDNA5 async-to-LDS builtin not available: using vector-load fallback")
  {
    const f32x4* g4 = (const f32x4*)g;
    f32x4* r4 = (f32x4*)row;
    for (int i = tid; i < C_DIM / 4; i += BLK1) r4[i] = g4[i];
  }
#endif
  __syncthreads();

  const f32x4* r4 = (const f32x4*)row;

  // ---- pass 1: row max ----
  float m = -__builtin_inff();
  for (int i = tid; i < C_DIM / 4; i += BLK1) {
    f32x4 v = r4[i];
    m = fmaxf(m, fmaxf(fmaxf(v.x, v.y), fmaxf(v.z, v.w)));
  }
  m = wave_max32(m);
  if (lane == 0) red[wave] = m;
  __syncthreads();
  float rowmax = red[0];
#pragma unroll
  for (int w = 1; w < BLK1 / 32; ++w) rowmax = fmaxf(rowmax, red[w]);
  __syncthreads();   // all waves done reading red before reuse

  // ---- pass 2: sum exp(x - rowmax) ----
  float s = 0.0f;
  for (int i = tid; i < C_DIM / 4; i += BLK1) {
    f32x4 v = r4[i];
    s += __expf(v.x - rowmax) + __expf(v.y - rowmax) +
         __expf(v.z - rowmax) + __expf(v.w - rowmax);
  }
  s = wave_sum32(s);
  if (lane == 0) red[wave] = s;
  __syncthreads();

  if (tid == 0) {
    float tot = 0.0f;
#pragma unroll
    for (int w = 0; w < BLK1 / 32; ++w) tot += red[w];
    const int t = tgt[rowid];           // target index within [0, C)
    const float xt = row[t];
    losses[rowid] = __logf(tot) + rowmax - xt;
  }
}

// ---------------------------------------------------------------------------
// Kernel 2: single workgroup. Bitonic-sort the 8192 losses ascending in LDS,
// then mean of the last KCOUNT entries (== mean of top-k; tie order is
// irrelevant to a top-k sum).
// ---------------------------------------------------------------------------
__global__ void __launch_bounds__(BLK2)
topk_mean_kernel(const float* __restrict__ losses, float* __restrict__ out) {
  __shared__ float arr[N_ROWS];        // 32 KB
  __shared__ float red[BLK2 / 32];
  const int tid  = threadIdx.x;
  const int lane = tid & 31;
  const int wave = tid >> 5;

  for (int i = tid; i < N_ROWS; i += BLK2) arr[i] = losses[i];
  __syncthreads();

  for (int k = 2; k <= N_ROWS; k <<= 1) {
    for (int j = k >> 1; j > 0; j >>= 1) {
      for (int i = tid; i < N_ROWS; i += BLK2) {
        const int ixj = i ^ j;
        if (ixj > i) {
          const float a = arr[i];
          const float b = arr[ixj];
          const bool up = ((i & k) == 0);
          if ((a > b) == up) { arr[i] = b; arr[ixj] = a; }
        }
      }
      __syncthreads();
    }
  }

  // ascending order: top-KCOUNT values live in arr[N_ROWS-KCOUNT .. N_ROWS-1]
  float s = 0.0f;
  for (int i = (N_ROWS - KCOUNT) + tid; i < N_ROWS; i += BLK2) s += arr[i];
  s = wave_sum32(s);
  if (lane == 0) red[wave] = s;
  __syncthreads();

  if (tid == 0) {
    float tot = 0.0f;
#pragma unroll
    for (int w = 0; w < BLK2 / 32; ++w) tot += red[w];
    out[0] = tot / (float)KCOUNT;
  }
}

// ---------------------------------------------------------------------------
extern "C" void kernel_launch(void* const* d_in, const int* in_sizes, int n_in,
                              void* d_out, int out_size, void* d_ws, size_t ws_size,
                              hipStream_t stream) {
  const float* inp = (const float*)d_in[0];   // [N, C] fp32
  const int*   tgt = (const int*)d_in[1];     // [N] int
  float* losses = (float*)d_ws;               // N floats of scratch
  float* out    = (float*)d_out;              // scalar fp32

  (void)in_sizes; (void)n_in; (void)out_size; (void)ws_size;

  row_nll_kernel<<<N_ROWS, BLK1, C_DIM * sizeof(float), stream>>>(inp, tgt, losses);
  topk_mean_kernel<<<1, BLK2, 0, stream>>>(losses, out);
}